// BUTDDetector_77506979824151
// MI455X (gfx1250) — compile-verified
//
#include <hip/hip_runtime.h>
#include <hip/hip_bf16.h>
#include <stdint.h>

// ---------------- problem constants ----------------
#define FH     512
#define FW     768
#define ANUM   12
#define HW     (FH * FW)                 // 393216
#define NTOT   (HW * ANUM)               // 4718592
#define PRE_N  6000
#define POST_N 300
#define SORT_N 8192
#define NEGV   (-1e30f)

// Anchor table (x1,y1 and w,h = x2-x1+1, y2-y1+1), faithful to generate_anchors(16,(.5,1,2),(4,8,16,32))
__constant__ float c_ax1[ANUM] = {-38.f,-84.f,-176.f,-360.f,-24.f,-56.f,-120.f,-248.f,-14.f,-36.f,-80.f,-168.f};
__constant__ float c_ay1[ANUM] = {-16.f,-40.f,-88.f,-184.f,-24.f,-56.f,-120.f,-248.f,-36.f,-80.f,-168.f,-344.f};
__constant__ float c_aw [ANUM] = { 92.f,184.f, 368.f, 736.f, 64.f,128.f, 256.f, 512.f, 44.f, 88.f,176.f, 352.f};
__constant__ float c_ah [ANUM] = { 48.f, 96.f, 192.f, 384.f, 64.f,128.f, 256.f, 512.f, 88.f,176.f,352.f, 704.f};

// ---------------- workspace layout (byte offsets) ----------------
#define KEYS_OFF   0u                      // u32[NTOT]
#define HIST_OFF   18874368u               // u32[256]
#define STATE_OFF  18875392u               // u32[8]: 0 prefix, 1 k_rem, 2 T, 3 need_eq, 4 cnt_sel, 5 cnt_eq
#define SELK_OFF   18875456u               // u32[PRE_N]
#define SELI_OFF   18899456u               // u32[PRE_N]
#define SIDX_OFF   18923456u               // u32[PRE_N]
#define BOX_OFF    18947456u               // float4[PRE_N] (16B aligned)
#define SCORE_OFF  19043456u               // f32[PRE_N]
#define SUP_OFF    19067456u               // u32[PRE_N]

typedef __attribute__((ext_vector_type(2))) float v2f;
typedef __attribute__((ext_vector_type(8))) float v8f;

// order-preserving float -> uint (larger float => larger key)
__device__ __forceinline__ unsigned key_of(float f) {
  unsigned u = __float_as_uint(f);
  return (u & 0x80000000u) ? ~u : (u | 0x80000000u);
}

struct Prop { float x1, y1, x2, y2, score; int valid; };

// i is in reference index order: i = loc*ANUM + a, loc = y*W + x
__device__ __forceinline__ Prop compute_prop(int i, const float* __restrict__ cls,
                                             const float* __restrict__ bbox,
                                             const float* __restrict__ im) {
  int loc = i / ANUM;
  int a   = i - loc * ANUM;
  int x   = loc % FW;
  int y   = loc / FW;

  float score = cls[(ANUM + a) * HW + loc];
  float d0 = bbox[(4 * a + 0) * HW + loc];
  float d1 = bbox[(4 * a + 1) * HW + loc];
  float d2 = bbox[(4 * a + 2) * HW + loc];
  float d3 = bbox[(4 * a + 3) * HW + loc];

  float w  = c_aw[a], h = c_ah[a];
  float cx = c_ax1[a] + 16.f * (float)x + 0.5f * w;
  float cy = c_ay1[a] + 16.f * (float)y + 0.5f * h;

  float pcx = d0 * w + cx;
  float pcy = d1 * h + cy;
  float pw  = expf(d2) * w;
  float ph  = expf(d3) * h;

  float him = im[0], wim = im[1], scl = im[2];
  float x1 = fminf(fmaxf(pcx - 0.5f * pw, 0.f), wim - 1.f);
  float y1 = fminf(fmaxf(pcy - 0.5f * ph, 0.f), him - 1.f);
  float x2 = fminf(fmaxf(pcx + 0.5f * pw, 0.f), wim - 1.f);
  float y2 = fminf(fmaxf(pcy + 0.5f * ph, 0.f), him - 1.f);

  float ms = 16.f * scl - 1.f;
  Prop p;
  p.x1 = x1; p.y1 = y1; p.x2 = x2; p.y2 = y2; p.score = score;
  p.valid = ((x2 - x1) >= ms) && ((y2 - y1) >= ms);
  return p;
}

// ---------------- kernel 0: init workspace state ----------------
__global__ void init_kernel(unsigned* hist, unsigned* state, unsigned* selk, unsigned* seli) {
  int t = threadIdx.x;
  for (int n = t; n < 256; n += 256) hist[n] = 0u;
  for (int n = t; n < 8; n += 256) state[n] = 0u;
  if (t == 0) state[1] = PRE_N;  // k_remaining
  for (int n = t; n < PRE_N; n += 256) { selk[n] = 0u; seli[n] = 0u; }
}

// ---------------- kernel 1: masked-score keys (coalesced j-order: j = a*HW + loc) ----------------
__global__ __launch_bounds__(256) void keys_kernel(const float* __restrict__ cls,
                                                   const float* __restrict__ bbox,
                                                   const float* __restrict__ im,
                                                   unsigned* __restrict__ keys) {
  int j = blockIdx.x * blockDim.x + threadIdx.x;
  if (j >= NTOT) return;
  int a   = j / HW;
  int loc = j - a * HW;
  int i   = loc * ANUM + a;                 // reference index order
  Prop p  = compute_prop(i, cls, bbox, im);
  float masked = p.valid ? p.score : NEGV;
  keys[j] = key_of(masked);
}

// ---------------- kernel 2: radix histogram (one of 4 passes) ----------------
__global__ __launch_bounds__(256) void hist_kernel(const unsigned* __restrict__ keys,
                                                   unsigned* __restrict__ hist,
                                                   const unsigned* __restrict__ state, int pass) {
  __shared__ unsigned lh[256];
  int t = threadIdx.x;
  lh[t] = 0u;
  __syncthreads();
  unsigned prefix = state[0];
  int shift = 8 * (3 - pass);
  int stride = gridDim.x * blockDim.x;
  for (int n = blockIdx.x * blockDim.x + t; n < NTOT; n += stride) {
    unsigned key = keys[n];
    bool match = (pass == 0) || ((key >> (shift + 8)) == (prefix >> (shift + 8)));
    if (match) atomicAdd(&lh[(key >> shift) & 0xFFu], 1u);
  }
  __syncthreads();
  if (lh[t]) atomicAdd(&hist[t], lh[t]);
}

// ---------------- kernel 3: WMMA 256-bin inclusive scan + digit select (1 wave) ----------------
// scan(16x16 X, row-major bins) = Lstrict*X*J + X*U, each 16x16x16 matmul = 4 chained v_wmma_f32_16x16x4_f32
__global__ __launch_bounds__(32) void scan_select_kernel(unsigned* __restrict__ hist,
                                                         unsigned* __restrict__ state, int pass) {
  __shared__ float histf[256];
  __shared__ float zbuf[256];
  __shared__ float sbuf[256];

  int l    = threadIdx.x;      // 0..31
  int half = l >> 4;           // 0/1
  int col  = l & 15;           // A row (M) and B/C/D column (N)

  for (int t = 0; t < 8; ++t) histf[l * 8 + t] = (float)hist[l * 8 + t];
  __syncthreads();

  // Y = X * U  (U upper-tri incl. diag)
  v8f accY = {0.f, 0.f, 0.f, 0.f, 0.f, 0.f, 0.f, 0.f};
  for (int k = 0; k < 4; ++k) {
    int kb = 4 * k + 2 * half;
    v2f a, b;
    a.x = histf[16 * col + kb];
    a.y = histf[16 * col + kb + 1];
    b.x = (float)(kb     <= col);
    b.y = (float)(kb + 1 <= col);
    accY = __builtin_amdgcn_wmma_f32_16x16x4_f32(false, a, false, b, (short)0, accY, false, false);
  }

  // Z = Lstrict * X
  v8f accZ = {0.f, 0.f, 0.f, 0.f, 0.f, 0.f, 0.f, 0.f};
  for (int k = 0; k < 4; ++k) {
    int kb = 4 * k + 2 * half;
    v2f a, b;
    a.x = (float)(kb     < col);
    a.y = (float)(kb + 1 < col);
    b.x = histf[16 * kb + col];
    b.y = histf[16 * (kb + 1) + col];
    accZ = __builtin_amdgcn_wmma_f32_16x16x4_f32(false, a, false, b, (short)0, accZ, false, false);
  }
  for (int v = 0; v < 8; ++v) zbuf[16 * (v + 8 * half) + col] = accZ[v];
  __syncthreads();

  // scan = Z * J + Y  (J = all-ones)
  v8f accF = accY;
  for (int k = 0; k < 4; ++k) {
    int kb = 4 * k + 2 * half;
    v2f a, b;
    a.x = zbuf[16 * col + kb];
    a.y = zbuf[16 * col + kb + 1];
    b.x = 1.f; b.y = 1.f;
    accF = __builtin_amdgcn_wmma_f32_16x16x4_f32(false, a, false, b, (short)0, accF, false, false);
  }
  for (int v = 0; v < 8; ++v) sbuf[16 * (v + 8 * half) + col] = accF[v];
  __syncthreads();

  if (l == 0) {
    float total  = sbuf[255];
    unsigned k_rem = state[1];
    float target = total - (float)k_rem;   // counts < 2^24, exact in f32
    int d = 255;
    for (int b = 0; b < 256; ++b) { if (sbuf[b] > target) { d = b; break; } }
    unsigned greater  = (unsigned)(total - sbuf[d] + 0.5f);
    unsigned k_new    = k_rem - greater;
    unsigned prefix   = state[0] | ((unsigned)d << (8 * (3 - pass)));
    state[0] = prefix;
    state[1] = k_new;
    if (pass == 3) { state[2] = prefix; state[3] = k_new; }  // T and need_eq
  }
  // clear histogram for next pass
  for (int t = 0; t < 8; ++t) hist[l * 8 + t] = 0u;
}

// ---------------- kernel 4: compact top-6000 (key > T, plus need_eq ties) ----------------
__global__ __launch_bounds__(256) void compact_kernel(const unsigned* __restrict__ keys,
                                                      unsigned* __restrict__ state,
                                                      unsigned* __restrict__ selk,
                                                      unsigned* __restrict__ seli) {
  unsigned T = state[2];
  unsigned need_eq = state[3];
  int stride = gridDim.x * blockDim.x;
  for (int n = blockIdx.x * blockDim.x + threadIdx.x; n < NTOT; n += stride) {
    unsigned key = keys[n];
    if (key < T) continue;
    int a   = n / HW;
    int loc = n - a * HW;
    unsigned iorig = (unsigned)(loc * ANUM + a);
    if (key > T) {
      unsigned s = atomicAdd(&state[4], 1u);
      if (s < PRE_N) { selk[s] = key; seli[s] = iorig; }
    } else {
      unsigned e = atomicAdd(&state[5], 1u);
      if (e < need_eq) {
        unsigned s = atomicAdd(&state[4], 1u);
        if (s < PRE_N) { selk[s] = key; seli[s] = iorig; }
      }
    }
  }
}

// ---------------- kernel 5: bitonic sort 8192 (desc key, asc idx) in 64KB LDS ----------------
__global__ __launch_bounds__(1024) void sort_kernel(const unsigned* __restrict__ selk,
                                                    const unsigned* __restrict__ seli,
                                                    unsigned* __restrict__ sidx) {
  __shared__ unsigned long long comp[SORT_N];
  int t = threadIdx.x;
  for (int n = t; n < SORT_N; n += 1024) {
    comp[n] = (n < PRE_N)
      ? (((unsigned long long)selk[n] << 32) | (unsigned long long)(0xFFFFFFFFu - seli[n]))
      : 0ull;
  }
  __syncthreads();
  for (int k = 2; k <= SORT_N; k <<= 1) {
    for (int j = k >> 1; j > 0; j >>= 1) {
      for (int n = t; n < SORT_N; n += 1024) {
        int ixj = n ^ j;
        if (ixj > n) {
          unsigned long long a = comp[n], b = comp[ixj];
          bool desc = ((n & k) == 0);
          if (desc ? (a < b) : (a > b)) { comp[n] = b; comp[ixj] = a; }
        }
      }
      __syncthreads();
    }
  }
  for (int n = t; n < PRE_N; n += 1024)
    sidx[n] = 0xFFFFFFFFu - (unsigned)(comp[n] & 0xFFFFFFFFull);
}

// ---------------- kernel 6: gather sorted 6000 -> clipped boxes + masked scores ----------------
__global__ __launch_bounds__(256) void gather_kernel(const unsigned* __restrict__ sidx,
                                                     const float* __restrict__ cls,
                                                     const float* __restrict__ bbox,
                                                     const float* __restrict__ im,
                                                     float4* __restrict__ boxes,
                                                     float* __restrict__ scores) {
  int j = blockIdx.x * blockDim.x + threadIdx.x;
  if (j >= PRE_N) return;
  int i = (int)sidx[j];
  Prop p = compute_prop(i, cls, bbox, im);
  boxes[j]  = make_float4(p.x1, p.y1, p.x2, p.y2);
  scores[j] = p.valid ? p.score : NEGV;
}

// ---------------- kernel 7: greedy NMS (boxes async-staged global->LDS) ----------------
__global__ __launch_bounds__(1024) void nms_kernel(const float4* __restrict__ boxes,
                                                   unsigned* __restrict__ sup_out) {
  __shared__ float4 box[PRE_N];          // 96 KB (WGP has 320 KB LDS)
  __shared__ unsigned char sup[PRE_N];
  int t = threadIdx.x;

  // CDNA5 async global->LDS staging: one B128 per box, tracked with ASYNCcnt
  for (int c = t; c < PRE_N; c += 1024) {
    unsigned lds_off = (unsigned)(size_t)(&box[c]);
    const float4* g = boxes + c;
    asm volatile("global_load_async_to_lds_b128 %0, %1, off"
                 :: "v"(lds_off), "v"(g) : "memory");
    sup[c] = 0;
  }
  asm volatile("s_wait_asynccnt 0" ::: "memory");
  __syncthreads();

  for (int i = 0; i < PRE_N; ++i) {
    if (!sup[i]) {                       // uniform branch (all lanes read same LDS byte)
      float4 bi = box[i];
      float area_i = (bi.z + 1.f - bi.x) * (bi.w + 1.f - bi.y);
      for (int j = i + 1 + t; j < PRE_N; j += 1024) {
        if (!sup[j]) {
          float4 bj = box[j];
          float xx1 = fmaxf(bi.x, bj.x), yy1 = fmaxf(bi.y, bj.y);
          float xx2 = fminf(bi.z, bj.z), yy2 = fminf(bi.w, bj.w);
          float w = fmaxf(0.f, xx2 + 1.f - xx1);
          float h = fmaxf(0.f, yy2 + 1.f - yy1);
          float inter = w * h;
          float area_j = (bj.z + 1.f - bj.x) * (bj.w + 1.f - bj.y);
          float iou = inter / (area_i + area_j - inter);
          if (iou > 0.7f) sup[j] = 1;
        }
      }
    }
    __syncthreads();
  }

  for (int c = t; c < PRE_N; c += 1024) sup_out[c] = (unsigned)sup[c];
}

// ---------------- kernel 8: take first 300 kept, write rois + scores ----------------
__global__ __launch_bounds__(512) void finalize_kernel(const unsigned* __restrict__ sup,
                                                       const float4* __restrict__ boxes,
                                                       const float* __restrict__ scores,
                                                       float* __restrict__ out) {
  __shared__ int list[POST_N];
  __shared__ int count;
  if (threadIdx.x == 0) {
    int n = 0;
    for (int j = 0; j < PRE_N && n < POST_N; ++j)
      if (!sup[j] && scores[j] > NEGV) list[n++] = j;
    count = n;
  }
  __syncthreads();
  for (int r = threadIdx.x; r < POST_N; r += blockDim.x) {
    if (r < count) {
      int j = list[r];
      float4 b = boxes[j];
      out[5 * r + 0] = 0.f;
      out[5 * r + 1] = b.x;
      out[5 * r + 2] = b.y;
      out[5 * r + 3] = b.z;
      out[5 * r + 4] = b.w;
      out[5 * POST_N + r] = scores[j];
    } else {
      out[5 * r + 0] = 0.f; out[5 * r + 1] = 0.f; out[5 * r + 2] = 0.f;
      out[5 * r + 3] = 0.f; out[5 * r + 4] = 0.f;
      out[5 * POST_N + r] = 0.f;
    }
  }
}

// ---------------- launch ----------------
extern "C" void kernel_launch(void* const* d_in, const int* in_sizes, int n_in,
                              void* d_out, int out_size, void* d_ws, size_t ws_size,
                              hipStream_t stream) {
  const float* cls  = (const float*)d_in[0];
  const float* bbox = (const float*)d_in[1];
  const float* im   = (const float*)d_in[2];
  float* out = (float*)d_out;

  char* ws = (char*)d_ws;
  unsigned* keys  = (unsigned*)(ws + KEYS_OFF);
  unsigned* hist  = (unsigned*)(ws + HIST_OFF);
  unsigned* state = (unsigned*)(ws + STATE_OFF);
  unsigned* selk  = (unsigned*)(ws + SELK_OFF);
  unsigned* seli  = (unsigned*)(ws + SELI_OFF);
  unsigned* sidx  = (unsigned*)(ws + SIDX_OFF);
  float4*   boxes = (float4*)(ws + BOX_OFF);
  float*    scores= (float*)(ws + SCORE_OFF);
  unsigned* supg  = (unsigned*)(ws + SUP_OFF);

  init_kernel<<<1, 256, 0, stream>>>(hist, state, selk, seli);
  keys_kernel<<<NTOT / 256, 256, 0, stream>>>(cls, bbox, im, keys);
  for (int p = 0; p < 4; ++p) {
    hist_kernel<<<1152, 256, 0, stream>>>(keys, hist, state, p);
    scan_select_kernel<<<1, 32, 0, stream>>>(hist, state, p);
  }
  compact_kernel<<<4608, 256, 0, stream>>>(keys, state, selk, seli);
  sort_kernel<<<1, 1024, 0, stream>>>(selk, seli, sidx);
  gather_kernel<<<(PRE_N + 255) / 256, 256, 0, stream>>>(sidx, cls, bbox, im, boxes, scores);
  nms_kernel<<<1, 1024, 0, stream>>>(boxes, supg);
  finalize_kernel<<<1, 512, 0, stream>>>(supg, boxes, scores, out);
}